// MPS_RNN_2D_8297876816277
// MI455X (gfx1250) — compile-verified
//
#include <hip/hip_runtime.h>
#include <hip/hip_bf16.h>

// ---------------------------------------------------------------------------
// 2-D MPS-RNN forward: per-step affine map done with V_WMMA_F32_16X16X4_F32.
// L=M=8, DCUT=6, HL=2, N=64 steps, B=32768 samples.
// Each wave32 owns 16 samples: D tile = 16 (padded ao rows, m=a*8+o) x 16 samples.
// Latency-bound serial scan -> runtime loop over 4 row pairs (16 steps unrolled
// inside, snake columns compile-time), double-buffered A fragments in registers,
// permlanex16 half swaps, native rsq/rcp/sqrt in the dependent chain, per-step
// scheduling fences to cap scalar liveness (avoid SGPR/VGPR spills).
// ---------------------------------------------------------------------------

#define LROWS 8
#define MCOLS 8
#define DCUT 6
#define HL 2
#define NSTEP 64
#define BATCH 32768
#define KFEAT 52              // 49 features padded to 52 (13 chunks of 4)
#define NCHUNK 13
#define A_CHUNK_FLOATS 64     // [half(2)][m(16)][r(2)]
#define A_TOTAL_FLOATS (NSTEP * NCHUNK * A_CHUNK_FLOATS)
#define SP_TOTAL_FLOATS (NSTEP * 16)   // also serves as prefetch overrun pad

typedef __attribute__((ext_vector_type(2))) float v2f;
typedef __attribute__((ext_vector_type(8))) float v8f;

// ------------------------- compile-time for loop ---------------------------
template <int I> struct IC { static constexpr int value = I; };

template <int N, typename F>
__device__ __forceinline__ void sfor(F&& f) {
  if constexpr (N > 0) {
    sfor<N - 1>((F&&)f);
    f(IC<N - 1>{});
  }
}

// xor-16 half swap within wave32: v_permlanex16_b32 with identity selects
__device__ __forceinline__ float swapx16(float v) {
  unsigned u = __builtin_bit_cast(unsigned, v);
  unsigned r = __builtin_amdgcn_permlanex16(u, u, 0x76543210u, 0xFEDCBA98u, false, false);
  return __builtin_bit_cast(float, r);
}

// feature f of the 52-long feature vector, from per-lane hh/hv registers
template <int F>
__device__ __forceinline__ float featf(const float (&hh)[DCUT], const float (&hv)[DCUT]) {
  if constexpr (F < 6)        return hh[F];
  else if constexpr (F < 12)  return hv[F - 6];
  else if constexpr (F < 48)  return hh[(F - 12) / 6] * hv[(F - 12) % 6];
  else if constexpr (F == 48) return 1.0f;
  else                        return 0.0f;
}

// ---------------------------------------------------------------------------
// Prep kernel: build per-step A matrices in WMMA A-fragment order plus the
// per-step scalar block {eta^2[6], w[6], c}.
// A layout: Apre[(k*13+c)*64 + half*32 + m*2 + r] = A_k[m][4c + 2*half + r]
//   rows m = a*8+o (o<6), zero otherwise.
//   cols f: [0..5]=Mh[a,o,:]+I, [6..11]=Mv[a,o,:]+I, [12..47]=T[a,o,c,d],
//           [48]=V[a,o], [49..51]=0.
// ---------------------------------------------------------------------------
__global__ void mps_prep_kernel(const float* __restrict__ Mh,
                                const float* __restrict__ Mv,
                                const float* __restrict__ V,
                                const float* __restrict__ T,
                                const float* __restrict__ W,
                                const float* __restrict__ Cc,
                                const float* __restrict__ Eta,
                                float* __restrict__ Apre,
                                float* __restrict__ StepP) {
  const int tid = blockIdx.x * blockDim.x + threadIdx.x;
  const int TOTAL_A = NSTEP * 16 * KFEAT;
  if (tid < TOTAL_A) {
    const int f = tid % KFEAT;
    const int m = (tid / KFEAT) % 16;
    const int k = tid / (KFEAT * 16);
    const int i = k >> 3;
    const int jm = k & 7;
    const int j = (i & 1) ? (7 - jm) : jm;           // snake traversal
    const int site = i * MCOLS + j;
    const int o = m & 7;
    const int a = m >> 3;
    float val = 0.0f;
    if (o < DCUT) {
      if (f < 6) {
        val = Mh[((site * HL + a) * DCUT + o) * DCUT + f] + (o == f ? 1.0f : 0.0f);
      } else if (f < 12) {
        const int d = f - 6;
        val = Mv[((site * HL + a) * DCUT + o) * DCUT + d] + (o == d ? 1.0f : 0.0f);
      } else if (f < 48) {
        const int cc = (f - 12) / 6;
        const int d  = (f - 12) % 6;
        val = T[(((site * HL + a) * DCUT + o) * DCUT + cc) * DCUT + d];
      } else if (f == 48) {
        val = V[(site * HL + a) * DCUT + o];
      }
    }
    const int c = f >> 2;
    const int h = (f >> 1) & 1;
    const int r = f & 1;
    Apre[(k * NCHUNK + c) * A_CHUNK_FLOATS + h * 32 + m * 2 + r] = val;
  } else {
    const int t2 = tid - TOTAL_A;
    if (t2 < NSTEP * 16) {
      const int idx = t2 & 15;
      const int k = t2 >> 4;
      const int i = k >> 3;
      const int jm = k & 7;
      const int j = (i & 1) ? (7 - jm) : jm;
      const int site = i * MCOLS + j;
      float v = 0.0f;
      if (idx < 6) {
        const float e = Eta[site * DCUT + idx];
        v = e * e;
      } else if (idx < 12) {
        v = W[site * DCUT + (idx - 6)];
      } else if (idx == 12) {
        v = Cc[site];
      }
      StepP[k * 16 + idx] = v;
    }
  }
}

// ---------------------------------------------------------------------------
// Main scan kernel. 256 threads = 8 waves per block; each wave owns 16
// samples. lane = tid&31, col = lane&15 (sample), half = lane>>4 (a index).
// Runtime loop over 4 row pairs; 16 steps unrolled per pair so snake columns
// are compile-time and v_state stays in registers. 13 chained f32 WMMAs per
// step; A fragments for step k+1 prefetched into a register double buffer.
// ---------------------------------------------------------------------------
__global__ __launch_bounds__(256)
void mps_rnn_main_kernel(const float* __restrict__ Apre,
                         const float* __restrict__ StepP,
                         const int* __restrict__ x,
                         float* __restrict__ out) {
  const int tid  = threadIdx.x;
  const int lane = tid & 31;
  const int col  = lane & 15;
  const int half = lane >> 4;
  const int wave = blockIdx.x * (blockDim.x >> 5) + (tid >> 5);
  const int s    = wave * 16 + col;   // sample index (same for both halves)

  // pack this sample's 64 occupation bits (b128 loads)
  unsigned long long bits = 0ull;
  {
    const int4* xp4 = (const int4*)(x + (long long)s * NSTEP);
#pragma unroll
    for (int w = 0; w < 16; ++w) {
      const int4 v = xp4[w];
      bits |= ((unsigned long long)(v.x & 1) << (4 * w + 0))
            | ((unsigned long long)(v.y & 1) << (4 * w + 1))
            | ((unsigned long long)(v.z & 1) << (4 * w + 2))
            | ((unsigned long long)(v.w & 1) << (4 * w + 3));
    }
  }

  float hh[DCUT], vst[MCOLS][DCUT];
#pragma unroll
  for (int j = 0; j < MCOLS; ++j)
#pragma unroll
    for (int o = 0; o < DCUT; ++o) vst[j][o] = 0.0f;   // bottom boundary
#pragma unroll
  for (int o = 0; o < DCUT; ++o) hh[o] = 1.0f;

  float amp = 1.0f, phi = 0.0f;
  const int aoff = half * 32 + col * 2;   // lane offset into a 64-float A chunk

  // double-buffered A fragments; preload global step 0
  v2f af[2][NCHUNK];
  sfor<NCHUNK>([&](auto CC) {
    constexpr int c = CC.value;
    af[0][c] = *(const v2f*)(Apre + c * A_CHUNK_FLOATS + aoff);
  });

  for (int rp = 0; rp < LROWS / 2; ++rp) {            // 4 row pairs
    const float* abase = Apre + (size_t)rp * (16 * NCHUNK * A_CHUNK_FLOATS);
    const float* spb   = StepP + rp * (16 * 16);
    const unsigned rowbits = (unsigned)(bits >> (rp * 16)) & 0xFFFFu;

    sfor<16>([&](auto KK) {
      constexpr int kk = KK.value;                    // step within row pair
      constexpr int j  = (kk < 8) ? kk : (15 - kk);   // snake column (const)
      constexpr int cb = kk & 1;                      // current A buffer
      constexpr int nb = cb ^ 1;                      // next A buffer

      // scheduling fence: stop cross-step hoisting of scalar loads
      asm volatile("" ::: "memory");

      // prefetch next step's A fragments (kk==15 -> next row pair's step 0;
      // on the last pair this reads the StepP pad and is never consumed)
      sfor<NCHUNK>([&](auto CC) {
        constexpr int c = CC.value;
        af[nb][c] = *(const v2f*)(abase +
            ((kk + 1) * NCHUNK + c) * A_CHUNK_FLOATS + aoff);
      });

      if constexpr (kk == 0 || kk == 8) {             // left boundary each row
#pragma unroll
        for (int o = 0; o < DCUT; ++o) hh[o] = 1.0f;
      }
      float hv[DCUT];
#pragma unroll
      for (int o = 0; o < DCUT; ++o) hv[o] = vst[j][o];

      v8f acc = {0.f, 0.f, 0.f, 0.f, 0.f, 0.f, 0.f, 0.f};
      sfor<NCHUNK>([&](auto CC) {
        constexpr int c = CC.value;
        // B fragment: lane -> feat[4c + 2*half + {0,1}][col]
        const float f0 = featf<4 * c + 0>(hh, hv);
        const float f1 = featf<4 * c + 1>(hh, hv);
        const float f2 = featf<4 * c + 2>(hh, hv);
        const float f3 = featf<4 * c + 3>(hh, hv);
        v2f bfrag;
        bfrag.x = half ? f2 : f0;
        bfrag.y = half ? f3 : f1;
        acc = __builtin_amdgcn_wmma_f32_16x16x4_f32(
            /*neg_a=*/false, af[cb][c], /*neg_b=*/false, bfrag,
            /*c_mod=*/(short)0, acc, /*reuse_a=*/false, /*reuse_b=*/false);
      });

      // acc[o] = h_full[a=half][o] for sample `col`; rows 6,7 are zero pad.
      const float* sp = spb + kk * 16;               // {eta2[6], w[6], c}

      float ss = 0.f, qr = 0.f;
#pragma unroll
      for (int o = 0; o < DCUT; ++o) {
        const float a2 = acc[o] * acc[o];
        ss += a2;
        qr += sp[o] * a2;    // raw q: prob is scale-invariant, norm cancels
      }
      const float ss_o = swapx16(ss);
      const float qr_o = swapx16(qr);
      const float inv  = __builtin_amdgcn_rsqf(ss + ss_o + 1e-12f);
      const float rqs  = __builtin_amdgcn_rcpf(qr + qr_o);

      const int  t    = (int)((rowbits >> kk) & 1u);
      const bool take = (t == half);
      const float ps  = (take ? qr : qr_o) * rqs;    // prob of selected a
      amp *= __builtin_amdgcn_sqrtf(ps);

      float hs[DCUT];
      float ph = sp[12];
#pragma unroll
      for (int o = 0; o < DCUT; ++o) {
        const float ao_other = swapx16(acc[o]);
        const float a_sel    = take ? acc[o] : ao_other;
        hs[o] = a_sel * inv;                         // normalized selected h
        ph += sp[6 + o] * hs[o];
      }
      phi += ph;

#pragma unroll
      for (int o = 0; o < DCUT; ++o) {
        hh[o] = hs[o];
        vst[j][o] = hs[o];
      }
    });
  }

  if (half == 0) {                                    // one writer per sample
    out[2 * (long long)s]     = amp * cosf(phi);
    out[2 * (long long)s + 1] = amp * sinf(phi);
  }
}

// ---------------------------------------------------------------------------
extern "C" void kernel_launch(void* const* d_in, const int* in_sizes, int n_in,
                              void* d_out, int out_size, void* d_ws, size_t ws_size,
                              hipStream_t stream) {
  const float* Mh  = (const float*)d_in[0];  // (8,8,2,6,6)
  const float* Mv  = (const float*)d_in[1];  // (8,8,2,6,6)
  const float* V   = (const float*)d_in[2];  // (8,8,2,6)
  const float* T   = (const float*)d_in[3];  // (8,8,2,6,6,6)
  const float* W   = (const float*)d_in[4];  // (8,8,6)
  const float* Cc  = (const float*)d_in[5];  // (8,8)
  const float* Eta = (const float*)d_in[6];  // (8,8,6)
  const int*   x   = (const int*)d_in[7];    // (B, 64)
  float* out = (float*)d_out;                // complex64 -> interleaved floats

  float* Apre  = (float*)d_ws;
  float* StepP = Apre + A_TOTAL_FLOATS;      // doubles as prefetch overrun pad

  const int prep_total = NSTEP * 16 * KFEAT + SP_TOTAL_FLOATS;
  mps_prep_kernel<<<(prep_total + 255) / 256, 256, 0, stream>>>(
      Mh, Mv, V, T, W, Cc, Eta, Apre, StepP);

  // 2048 waves x 16 samples = 32768; 8 waves per 256-thread block -> 256 blocks
  mps_rnn_main_kernel<<<BATCH / 128, 256, 0, stream>>>(Apre, StepP, x, out);
}